// MLS_23390391894261
// MI455X (gfx1250) — compile-verified
//
#include <hip/hip_runtime.h>

// Problem constants (B, C, H, W) = (4, 256, 256, 256), 68-17 = 51 landmarks.
#define NB   4
#define NC   256
#define NH   256
#define NW   256
#define NLM  51
#define NCHUNK 13   // ceil(51/4) K-chunks of 4 for WMMA f32 16x16x4

typedef __attribute__((ext_vector_type(2))) float v2f;
typedef __attribute__((ext_vector_type(8))) float v8f;

// ---------------------------------------------------------------------------
// Stats -> sample coordinate.
// S = [wsum, Sp0, Sp1, Sq0, Sq1, Mpq00, Mpq01, Mpq10, Mpq11, Mpp00, Mpp01, Mpp11]
// pTwq_ij = Mpq_ij - Sp_i*Sq_j/wsum ; pTwp_ij = Mpp_ij - Sp_i*Sp_j/wsum
// ---------------------------------------------------------------------------
__device__ __forceinline__ float2 stats_to_xy(const float* S, float hF, float wF) {
    float inv_wsum = 1.0f / S[0];
    float ps0 = S[1] * inv_wsum, ps1 = S[2] * inv_wsum;
    float qs0 = S[3] * inv_wsum, qs1 = S[4] * inv_wsum;
    float A00 = S[5]  - S[1] * qs0;
    float A01 = S[6]  - S[1] * qs1;
    float A10 = S[7]  - S[2] * qs0;
    float A11 = S[8]  - S[2] * qs1;
    float P00 = S[9]  - S[1] * ps0;
    float P01 = S[10] - S[1] * ps1;
    float P11 = S[11] - S[2] * ps1;
    float det = A00 * A11 - A01 * A10;
    float invdet = 1.0f / det;
    float ml0 = hF - qs0, ml1 = wF - qs1;
    float a0 = ( ml0 * A11 - ml1 * A10) * invdet;
    float a1 = ( ml1 * A00 - ml0 * A01) * invdet;
    float t0 = a0 * P00 + a1 * P01 + ps0;   // row coordinate
    float t1 = a0 * P01 + a1 * P11 + ps1;   // col coordinate
    t0 = fmaxf(t0, 0.0f);
    t1 = fmaxf(t1, 0.0f);
    t0 = (t0 > (float)(NH - 1)) ? 0.0f : t0;
    t1 = (t1 > (float)(NW - 1)) ? 0.0f : t1;
    // grid_sample algebra collapses to x = t1 - 0.5, y = t0 - 0.5
    return make_float2(t1 - 0.5f, t0 - 0.5f);
}

// ---------------------------------------------------------------------------
// Bilinear sample of all 256 channels at one pixel. Weights/addresses hoisted
// out of the channel loop; lanes cover consecutive w -> coalesced stores.
// ---------------------------------------------------------------------------
__device__ __forceinline__ void sample_pixel(const float* __restrict__ fea,
                                             float* __restrict__ out,
                                             int b, int h, int w,
                                             float x, float y) {
    float x0f = floorf(x), y0f = floorf(y);
    float wx1 = x - x0f,   wy1 = y - y0f;
    float wx0 = 1.0f - wx1, wy0 = 1.0f - wy1;
    float x1f = x0f + 1.0f, y1f = y0f + 1.0f;
    float vx0 = (x0f >= 0.0f && x0f <= (float)(NW - 1)) ? 1.0f : 0.0f;
    float vx1 = (x1f >= 0.0f && x1f <= (float)(NW - 1)) ? 1.0f : 0.0f;
    float vy0 = (y0f >= 0.0f && y0f <= (float)(NH - 1)) ? 1.0f : 0.0f;
    float vy1 = (y1f >= 0.0f && y1f <= (float)(NH - 1)) ? 1.0f : 0.0f;
    int xi0 = min(max((int)x0f, 0), NW - 1);
    int xi1 = min(max((int)x1f, 0), NW - 1);
    int yi0 = min(max((int)y0f, 0), NH - 1);
    int yi1 = min(max((int)y1f, 0), NH - 1);
    float w00 = wy0 * wx0 * vy0 * vx0;
    float w01 = wy0 * wx1 * vy0 * vx1;
    float w10 = wy1 * wx0 * vy1 * vx0;
    float w11 = wy1 * wx1 * vy1 * vx1;
    int o00 = (yi0 << 8) + xi0;
    int o01 = (yi0 << 8) + xi1;
    int o10 = (yi1 << 8) + xi0;
    int o11 = (yi1 << 8) + xi1;
    const float* fb = fea + ((size_t)b << 24);                 // b * C*H*W (2^24)
    float*       ob = out + ((size_t)b << 24) + (h << 8) + w;
#pragma unroll 4
    for (int c = 0; c < NC; ++c) {
        int cb = c << 16;                                      // c * H*W
        float v = fb[cb + o00] * w00 + fb[cb + o01] * w01 +
                  fb[cb + o10] * w10 + fb[cb + o11] * w11;
        ob[cb] = v;
    }
}

// ---------------------------------------------------------------------------
// Kernel 1: MLS grid via WMMA. Each wave32 computes 16 pixels:
//   D(16 pix x 12 stats) = A(16 pix x 52 lm weights) * B(52 lm x 12 stat feats)
// as 13 V_WMMA_F32_16X16X4_F32 accumulations. Landmark 51 is a zero B-row.
// ---------------------------------------------------------------------------
__global__ __launch_bounds__(256) void mls_grid_wmma(const float* __restrict__ Ld,
                                                     const float* __restrict__ Lg,
                                                     float2* __restrict__ xy) {
    __shared__ float Bs[NCHUNK][4][16];   // [chunk][k-row][stat]
    __shared__ float Ds[8][16][17];       // [wave][pixel][stat], padded

    const int blockBase = blockIdx.x * 128;        // 128 pixels per block
    const int b = blockBase >> 16;                 // uniform per block
    const int tid = threadIdx.x;

    // Build B (landmark feature matrix) cooperatively: 13*4*16 = 832 entries.
    for (int e = tid; e < NCHUNK * 64; e += 256) {
        int c = e >> 6, r = (e >> 4) & 3, n = e & 15;
        int k = c * 4 + r;
        float val = 0.0f;
        if (k < NLM) {
            int base = (b * 68 + 17 + k) * 2;
            float P0 = Lg[base + 1], P1 = Lg[base + 0];   // p[:, ::-1]
            float Q0 = Ld[base + 1], Q1 = Ld[base + 0];
            switch (n) {
                case 0:  val = 1.0f;    break;
                case 1:  val = P0;      break;
                case 2:  val = P1;      break;
                case 3:  val = Q0;      break;
                case 4:  val = Q1;      break;
                case 5:  val = P0 * Q0; break;
                case 6:  val = P0 * Q1; break;
                case 7:  val = P1 * Q0; break;
                case 8:  val = P1 * Q1; break;
                case 9:  val = P0 * P0; break;
                case 10: val = P0 * P1; break;
                case 11: val = P1 * P1; break;
                default: val = 0.0f;    break;
            }
        }
        Bs[c][r][n] = val;
    }
    __syncthreads();

    const int wave = tid >> 5;
    const int lane = tid & 31;
    const int m    = lane & 15;        // pixel row of A / stat column of B
    const int half = lane >> 4;        // K split: rows {0,1} vs {2,3}
    const int pixelBase = blockBase + wave * 16;
    const float hF = (float)((pixelBase >> 8) & 255);
    const float wF = (float)((pixelBase & 255) + m);
    const int r0 = half * 2, r1 = r0 + 1;

    v8f acc = {0.f, 0.f, 0.f, 0.f, 0.f, 0.f, 0.f, 0.f};
#pragma unroll
    for (int c = 0; c < NCHUNK; ++c) {
        // A fragment: two MLS weights for this lane's pixel (K rows r0, r1).
        float P0a = Bs[c][r0][1], P1a = Bs[c][r0][2];
        float P0b = Bs[c][r1][1], P1b = Bs[c][r1][2];
        float d0a = P0a - hF, d1a = P1a - wF;
        float d0b = P0b - hF, d1b = P1b - wF;
        v2f a, bf;
        a.x = 1.0f / (d0a * d0a + d1a * d1a + 1e-8f);
        a.y = 1.0f / (d0b * d0b + d1b * d1b + 1e-8f);
        // B fragment: same K rows, stat column n = lane&15 (zero pad row k=51).
        bf.x = Bs[c][r0][m];
        bf.y = Bs[c][r1][m];
        acc = __builtin_amdgcn_wmma_f32_16x16x4_f32(false, a, false, bf,
                                                    (short)0, acc, false, false);
    }

    // D layout: VGPR r -> rows r (lanes 0-15) / r+8 (lanes 16-31), col = lane&15.
#pragma unroll
    for (int r = 0; r < 8; ++r)
        Ds[wave][r + half * 8][m] = acc[r];
    __syncthreads();

    if (tid < 128) {
        int p  = tid;
        int pf = blockBase + p;
        float S[12];
#pragma unroll
        for (int j = 0; j < 12; ++j) S[j] = Ds[p >> 4][p & 15][j];
        float hh = (float)((pf >> 8) & 255);
        float ww = (float)(pf & 255);
        xy[pf] = stats_to_xy(S, hh, ww);
    }
}

// ---------------------------------------------------------------------------
// Kernel 2: bandwidth-bound bilinear gather (one thread = one pixel, all C).
// ---------------------------------------------------------------------------
__global__ __launch_bounds__(256) void mls_sample(const float* __restrict__ fea,
                                                  const float2* __restrict__ xy,
                                                  float* __restrict__ out) {
    int pf = blockIdx.x * 256 + threadIdx.x;
    int b = pf >> 16, h = (pf >> 8) & 255, w = pf & 255;
    float2 g = xy[pf];
    sample_pixel(fea, out, b, h, w, g.x, g.y);
}

// ---------------------------------------------------------------------------
// Fallback: fused scalar MLS + sample (used only if workspace is too small).
// ---------------------------------------------------------------------------
__global__ __launch_bounds__(256) void mls_fused(const float* __restrict__ fea,
                                                 const float* __restrict__ Ld,
                                                 const float* __restrict__ Lg,
                                                 float* __restrict__ out) {
    int pf = blockIdx.x * 256 + threadIdx.x;
    int b = pf >> 16, h = (pf >> 8) & 255, w = pf & 255;
    float hF = (float)h, wF = (float)w;
    float S[12] = {0.f, 0.f, 0.f, 0.f, 0.f, 0.f, 0.f, 0.f, 0.f, 0.f, 0.f, 0.f};
    for (int k = 0; k < NLM; ++k) {
        int base = (b * 68 + 17 + k) * 2;
        float P0 = Lg[base + 1], P1 = Lg[base + 0];
        float Q0 = Ld[base + 1], Q1 = Ld[base + 0];
        float d0 = P0 - hF, d1 = P1 - wF;
        float wk = 1.0f / (d0 * d0 + d1 * d1 + 1e-8f);
        S[0] += wk;
        S[1] += wk * P0;      S[2]  += wk * P1;
        S[3] += wk * Q0;      S[4]  += wk * Q1;
        S[5] += wk * P0 * Q0; S[6]  += wk * P0 * Q1;
        S[7] += wk * P1 * Q0; S[8]  += wk * P1 * Q1;
        S[9] += wk * P0 * P0; S[10] += wk * P0 * P1; S[11] += wk * P1 * P1;
    }
    float2 g = stats_to_xy(S, hF, wF);
    sample_pixel(fea, out, b, h, w, g.x, g.y);
}

extern "C" void kernel_launch(void* const* d_in, const int* in_sizes, int n_in,
                              void* d_out, int out_size, void* d_ws, size_t ws_size,
                              hipStream_t stream) {
    const float* fea = (const float*)d_in[0];   // fea_g (B,C,H,W) f32
    const float* Ld  = (const float*)d_in[1];   // L_d   (B,68,2)  f32
    const float* Lg  = (const float*)d_in[2];   // L_g   (B,68,2)  f32
    float* out = (float*)d_out;
    const int npix = NB * NH * NW;              // 262144

    if (ws_size >= (size_t)npix * sizeof(float2)) {
        mls_grid_wmma<<<npix / 128, 256, 0, stream>>>(Ld, Lg, (float2*)d_ws);
        mls_sample<<<npix / 256, 256, 0, stream>>>(fea, (const float2*)d_ws, out);
    } else {
        mls_fused<<<npix / 256, 256, 0, stream>>>(fea, Ld, Lg, out);
    }
}